// NodeRegressionGNN_58067957842223
// MI455X (gfx1250) — compile-verified
//
#include <hip/hip_runtime.h>
#include <hip/hip_bf16.h>

typedef float v2f __attribute__((ext_vector_type(2)));
typedef float v8f __attribute__((ext_vector_type(8)));

// ---------------------------------------------------------------------------
// Degree / normalization
// ---------------------------------------------------------------------------
__global__ __launch_bounds__(256) void deg_init(float* deg, int n) {
  int i = blockIdx.x * 256 + threadIdx.x;
  if (i < n) deg[i] = 1.0f;  // self-loop
}

__global__ __launch_bounds__(256) void deg_accum(const int* __restrict__ dst,
                                                 float* __restrict__ deg, int e) {
  int i = blockIdx.x * 256 + threadIdx.x;
  if (i < e) unsafeAtomicAdd(&deg[dst[i]], 1.0f);
}

__global__ __launch_bounds__(256) void deg_rsqrt(float* deg, int n) {
  int i = blockIdx.x * 256 + threadIdx.x;
  if (i < n) deg[i] = rsqrtf(deg[i]);
}

// Pad x [n,3] -> xpad [n,4] with zero in lane 3 (removes all K guards in GEMM)
__global__ __launch_bounds__(256) void pad_x(const float* __restrict__ x,
                                             float* __restrict__ xp, int n) {
  int i = blockIdx.x * 256 + threadIdx.x;
  if (i >= n) return;
  float4 v;
  v.x = x[i * 3 + 0];
  v.y = x[i * 3 + 1];
  v.z = x[i * 3 + 2];
  v.w = 0.0f;
  reinterpret_cast<float4*>(xp)[i] = v;
}

// ---------------------------------------------------------------------------
// fp32 WMMA GEMM: T[n,DOUT] = H[n,DIN] @ W[DIN,DOUT] (+bias, +relu)
// V_WMMA_F32_16X16X4_F32. Each wave: 2 row tiles (M=32) x 1 col tile (N=16),
// B fragment reused by both WMMAs per K-step. Block = 128 = 4 waves -> 4
// consecutive column tiles. n must be a multiple of 32 (100000 = 3125*32).
// Fully unrolled K loop; no guards, no divergence.
// ---------------------------------------------------------------------------
template <int DIN, int DOUT, bool BIAS, bool RELU>
__global__ __launch_bounds__(128)
void gemm_wmma(const float* __restrict__ H, const float* __restrict__ W,
               const float* __restrict__ bias, float* __restrict__ T) {
  const int lane = threadIdx.x & 31;
  const int wid  = threadIdx.x >> 5;
  const int half = lane >> 4;   // 0: lanes 0-15, 1: lanes 16-31
  const int l16  = lane & 15;
  const int tile_m = blockIdx.x * 32;
  const int tile_n = (blockIdx.y * 4 + wid) * 16;
  if (tile_n >= DOUT) return;   // wave-uniform (EXEC all-ones inside)
  const int row0 = tile_m + l16;
  const int col  = tile_n + l16;

  v8f c0 = {};
  v8f c1 = {};
#pragma unroll
  for (int ks = 0; ks < DIN / 4; ++ks) {
    const int k0 = ks * 4 + half * 2;
    // B 4x16: vgpr0 holds K = {0|2}, vgpr1 holds K = {1|3}
    v2f b;
    b.x = W[(k0 + 0) * DOUT + col];
    b.y = W[(k0 + 1) * DOUT + col];
    // A 16x4: lanes0-15 K={0,1}, lanes16-31 K={2,3}; contiguous -> b64 load
    v2f a0 = *reinterpret_cast<const v2f*>(H + (size_t)row0 * DIN + k0);
    v2f a1 = *reinterpret_cast<const v2f*>(H + (size_t)(row0 + 16) * DIN + k0);
    c0 = __builtin_amdgcn_wmma_f32_16x16x4_f32(false, a0, false, b, (short)0,
                                               c0, false, false);
    c1 = __builtin_amdgcn_wmma_f32_16x16x4_f32(false, a1, false, b, (short)0,
                                               c1, false, false);
  }
  float bv;
  if constexpr (BIAS) bv = bias[col];
#pragma unroll
  for (int r = 0; r < 8; ++r) {
    // C layout: vgpr r -> M = r (lanes 0-15) / r+8 (lanes 16-31), N = l16
    int orow = tile_m + r + half * 8;
    float v0 = c0[r];
    float v1 = c1[r];
    if constexpr (BIAS) { v0 += bv; v1 += bv; }
    if constexpr (RELU) { v0 = fmaxf(v0, 0.0f); v1 = fmaxf(v1, 0.0f); }
    T[(size_t)orow * DOUT + col]        = v0;
    T[(size_t)(orow + 16) * DOUT + col] = v1;
  }
}

// ---------------------------------------------------------------------------
// out[i,:] = bias + t[i,:] * dinv[i]^2   (self-loop contribution, atomic-free)
// ---------------------------------------------------------------------------
template <int D>
__global__ __launch_bounds__(256)
void agg_init(const float* __restrict__ t, const float* __restrict__ dinv,
              const float* __restrict__ bias, float* __restrict__ out, int n) {
  constexpr int Q = D >> 2;
  int i = blockIdx.x * 256 + threadIdx.x;
  if (i >= n * Q) return;
  int node = i / Q;            // Q is a power of two -> shift
  int c4 = (i & (Q - 1)) << 2;
  float dv = dinv[node];
  float s = dv * dv;
  float4 tv = reinterpret_cast<const float4*>(t)[i];
  float4 o;
  o.x = bias[c4 + 0] + tv.x * s;
  o.y = bias[c4 + 1] + tv.y * s;
  o.z = bias[c4 + 2] + tv.z * s;
  o.w = bias[c4 + 3] + tv.w * s;
  reinterpret_cast<float4*>(out)[i] = o;
}

// ---------------------------------------------------------------------------
// Per-edge scatter-add: out[dst] += t[src] * dinv[src]*dinv[dst]
// D/4 threads per edge, float4 gather, 4 f32 hardware atomics (L2-resident).
// ---------------------------------------------------------------------------
template <int D>
__global__ __launch_bounds__(256)
void edge_scatter(const int* __restrict__ src, const int* __restrict__ dst,
                  const float* __restrict__ t, const float* __restrict__ dinv,
                  float* __restrict__ out, int e) {
  constexpr int PER = D >> 2;
  int gid = blockIdx.x * 256 + threadIdx.x;
  int eid = gid / PER;         // power of two -> shift
  if (eid >= e) return;
  int c4 = (gid & (PER - 1)) << 2;
  int s  = src[eid];
  int dd = dst[eid];
  float nrm = dinv[s] * dinv[dd];
  float4 v = *reinterpret_cast<const float4*>(t + (size_t)s * D + c4);
  float* o = out + (size_t)dd * D + c4;
  unsafeAtomicAdd(o + 0, v.x * nrm);
  unsafeAtomicAdd(o + 1, v.y * nrm);
  unsafeAtomicAdd(o + 2, v.z * nrm);
  unsafeAtomicAdd(o + 3, v.w * nrm);
}

// ---------------------------------------------------------------------------
// In-place LayerNorm + ReLU, one wave32 per node, shuffle reduction.
// ---------------------------------------------------------------------------
template <int D>
__global__ __launch_bounds__(256)
void ln_relu(float* __restrict__ h, const float* __restrict__ g,
             const float* __restrict__ be, int n) {
  constexpr int PER = D >> 5;  // 2 or 4
  int node = blockIdx.x * 8 + (threadIdx.x >> 5);
  int lane = threadIdx.x & 31;
  if (node >= n) return;
  float* p = h + (size_t)node * D;
  float vals[PER];
  float s = 0.0f, ss = 0.0f;
#pragma unroll
  for (int j = 0; j < PER; ++j) {
    float v = p[lane + j * 32];
    vals[j] = v;
    s += v;
    ss += v * v;
  }
#pragma unroll
  for (int off = 16; off > 0; off >>= 1) {
    s  += __shfl_xor(s, off, 32);
    ss += __shfl_xor(ss, off, 32);
  }
  constexpr float inv_d = 1.0f / (float)D;
  float m = s * inv_d;
  float var = ss * inv_d - m * m;
  float rs = rsqrtf(var + 1e-5f);
#pragma unroll
  for (int j = 0; j < PER; ++j) {
    int cc = lane + j * 32;
    float v = (vals[j] - m) * rs * g[cc] + be[cc];
    p[cc] = fmaxf(v, 0.0f);
  }
}

// ---------------------------------------------------------------------------
// Final 32 -> 1 dot product, one wave per node.
// ---------------------------------------------------------------------------
__global__ __launch_bounds__(256)
void mlp_out(const float* __restrict__ h, const float* __restrict__ w,
             const float* __restrict__ b, float* __restrict__ out, int n) {
  int node = blockIdx.x * 8 + (threadIdx.x >> 5);
  int lane = threadIdx.x & 31;
  if (node >= n) return;
  float v = h[(size_t)node * 32 + lane] * w[lane];
#pragma unroll
  for (int off = 16; off > 0; off >>= 1) v += __shfl_xor(v, off, 32);
  if (lane == 0) out[node] = v + b[0];
}

// ---------------------------------------------------------------------------
static inline int cdiv(long long a, long long b) { return (int)((a + b - 1) / b); }

extern "C" void kernel_launch(void* const* d_in, const int* in_sizes, int n_in,
                              void* d_out, int out_size, void* d_ws, size_t ws_size,
                              hipStream_t stream) {
  const float* x   = (const float*)d_in[0];
  const int*   ei  = (const int*)d_in[1];
  const float* W0  = (const float*)d_in[2];
  const float* b0  = (const float*)d_in[3];
  const float* g0  = (const float*)d_in[4];
  const float* be0 = (const float*)d_in[5];
  const float* W1  = (const float*)d_in[6];
  const float* b1  = (const float*)d_in[7];
  const float* g1  = (const float*)d_in[8];
  const float* be1 = (const float*)d_in[9];
  const float* W2  = (const float*)d_in[10];
  const float* b2  = (const float*)d_in[11];
  const float* g2  = (const float*)d_in[12];
  const float* be2 = (const float*)d_in[13];
  const float* mW0 = (const float*)d_in[14];
  const float* mb0 = (const float*)d_in[15];
  const float* mW1 = (const float*)d_in[16];
  const float* mb1 = (const float*)d_in[17];
  const float* mW2 = (const float*)d_in[18];
  const float* mb2 = (const float*)d_in[19];
  float* out = (float*)d_out;

  const int n = in_sizes[0] / 3;   // 100000
  const int e = in_sizes[1] / 2;   // 1600000
  const int* srcI = ei;
  const int* dstI = ei + e;

  float* dinv = (float*)d_ws;                       // n floats
  float* xpad = dinv + n;                           // n*4 floats
  float* bufA = xpad + (size_t)n * 4;               // n*128 floats
  float* bufB = bufA + (size_t)n * 128;             // n*128 floats

  // --- degree -> dinv ; pad x ---
  deg_init <<<cdiv(n, 256), 256, 0, stream>>>(dinv, n);
  deg_accum<<<cdiv(e, 256), 256, 0, stream>>>(dstI, dinv, e);
  deg_rsqrt<<<cdiv(n, 256), 256, 0, stream>>>(dinv, n);
  pad_x    <<<cdiv(n, 256), 256, 0, stream>>>(x, xpad, n);

  const int mtiles = n / 32;  // 3125

  // --- GCN layer 0: (3->pad 4) -> 64 ---
  gemm_wmma<4, 64, false, false><<<dim3(mtiles, 1), 128, 0, stream>>>(xpad, W0, nullptr, bufA);
  agg_init<64>    <<<cdiv((long long)n * 16, 256), 256, 0, stream>>>(bufA, dinv, b0, bufB, n);
  edge_scatter<64><<<cdiv((long long)e * 16, 256), 256, 0, stream>>>(srcI, dstI, bufA, dinv, bufB, e);
  ln_relu<64>     <<<cdiv(n, 8), 256, 0, stream>>>(bufB, g0, be0, n);

  // --- GCN layer 1: 64 -> 128 ---
  gemm_wmma<64, 128, false, false><<<dim3(mtiles, 2), 128, 0, stream>>>(bufB, W1, nullptr, bufA);
  agg_init<128>    <<<cdiv((long long)n * 32, 256), 256, 0, stream>>>(bufA, dinv, b1, bufB, n);
  edge_scatter<128><<<cdiv((long long)e * 32, 256), 256, 0, stream>>>(srcI, dstI, bufA, dinv, bufB, e);
  ln_relu<128>     <<<cdiv(n, 8), 256, 0, stream>>>(bufB, g1, be1, n);

  // --- GCN layer 2: 128 -> 128 ---
  gemm_wmma<128, 128, false, false><<<dim3(mtiles, 2), 128, 0, stream>>>(bufB, W2, nullptr, bufA);
  agg_init<128>    <<<cdiv((long long)n * 32, 256), 256, 0, stream>>>(bufA, dinv, b2, bufB, n);
  edge_scatter<128><<<cdiv((long long)e * 32, 256), 256, 0, stream>>>(srcI, dstI, bufA, dinv, bufB, e);
  ln_relu<128>     <<<cdiv(n, 8), 256, 0, stream>>>(bufB, g2, be2, n);

  // --- MLP head ---
  gemm_wmma<128, 128, true, true><<<dim3(mtiles, 2), 128, 0, stream>>>(bufB, mW0, mb0, bufA);
  gemm_wmma<128, 32, true, true> <<<dim3(mtiles, 1), 128, 0, stream>>>(bufA, mW1, mb1, bufB);
  mlp_out<<<cdiv(n, 8), 256, 0, stream>>>(bufB, mW2, mb2, out, n);
}